// FedformerEncoder_39444979647100
// MI455X (gfx1250) — compile-verified
//
#include <hip/hip_runtime.h>

#define LTOT 8192
#define DCH  128

typedef __attribute__((ext_vector_type(2))) float v2f;
typedef __attribute__((ext_vector_type(8))) float v8f;

__device__ __forceinline__ float waveRedSum(float v) {
  v += __shfl_xor(v, 16, 32);
  v += __shfl_xor(v, 8, 32);
  v += __shfl_xor(v, 4, 32);
  v += __shfl_xor(v, 2, 32);
  v += __shfl_xor(v, 1, 32);
  return v;
}

// ---------------------------------------------------------------------------
// Kernel 1: series decomposition (25-tap moving average, zero-padded edges).
// Writes trend (row-major), seasonal row-major, seasonal channel-major [B,D,L].
// ---------------------------------------------------------------------------
__global__ __launch_bounds__(256) void decomp_kernel(
    const float* __restrict__ x, float* __restrict__ trend,
    float* __restrict__ seasR, float* __restrict__ seasC) {
  extern __shared__ float sm[];
  float* xt = sm;              // 88 * 128 tile (64 outputs + 12 halo each side)
  float* st = sm + 88 * 128;   // 64 * 129 seasonal tile (padded pitch)
  const int tid = threadIdx.x;
  const int l0 = blockIdx.x * 64;
  const int b = blockIdx.y;
  const size_t baseRow = ((size_t)b * LTOT + l0) * DCH;

  for (int i = tid; i < 88 * 128; i += 256) {
    int li = i >> 7, d = i & 127;
    int gl = l0 + li - 12;
    float v = 0.0f;
    if (gl >= 0 && gl < LTOT) v = x[((size_t)b * LTOT + gl) * DCH + d];
    xt[i] = v;
  }
  __syncthreads();

  for (int i = tid; i < 64 * 128; i += 256) {
    int l = i >> 7, d = i & 127;
    float s = 0.0f;
#pragma unroll
    for (int j = 0; j < 25; ++j) s += xt[(l + j) * 128 + d];
    float tr = s * (1.0f / 25.0f);
    float se = xt[(l + 12) * 128 + d] - tr;
    trend[baseRow + (size_t)l * DCH + d] = tr;
    seasR[baseRow + (size_t)l * DCH + d] = se;
    st[l * 129 + d] = se;
  }
  __syncthreads();

  for (int i = tid; i < 64 * 128; i += 256) {
    int d = i >> 6, l = i & 63;
    seasC[((size_t)b * DCH + d) * LTOT + l0 + l] = st[l * 129 + d];
  }
}

// ---------------------------------------------------------------------------
// Kernel 2: per-signal 8192-pt FFT in LDS, top-16 |X_k| selection over the
// 4097 rfft bins, sparse irfft reconstruction. One workgroup per (b, d).
// ---------------------------------------------------------------------------
__global__ __launch_bounds__(256) void fourier_kernel(
    const float* __restrict__ seasC, float* __restrict__ fsC) {
  extern __shared__ float sm[];
  float* re = sm;            // 8192
  float* im = sm + 8192;     // 8192
  float* twr = sm + 16384;   // 4096: cos(2*pi*q/8192)
  float* twi = sm + 20480;   // 4096: -sin(2*pi*q/8192)
  __shared__ float redv[256];
  __shared__ int redi[256];
  __shared__ int selk[16];
  __shared__ float selR[16], selI[16], selW[16];

  const int tid = threadIdx.x;
  const size_t sig = (size_t)blockIdx.x * LTOT;

  // bit-reversed load (coalesced global read, scattered LDS write)
  for (int i = tid; i < 8192; i += 256) {
    int r = (int)(__brev((unsigned)i) >> 19);
    re[r] = seasC[sig + i];
    im[r] = 0.0f;
  }
  // twiddle table tw[q] = exp(-2*pi*i*q/8192)
  for (int q = tid; q < 4096; q += 256) {
    float sn, cs;
    __sincosf(-6.28318530717958647692f * (float)q * (1.0f / 8192.0f), &sn, &cs);
    twr[q] = cs;
    twi[q] = sn;
  }
  __syncthreads();

  // 13 radix-2 DIT stages, natural-order output
  for (int s = 1; s <= 13; ++s) {
    const int hf = 1 << (s - 1);
    for (int idx = tid; idx < 4096; idx += 256) {
      int j = idx & (hf - 1);
      int p0 = ((idx >> (s - 1)) << s) + j;
      int p1 = p0 + hf;
      int q = j << (13 - s);
      float wr = twr[q], wi = twi[q];
      float ar = re[p0], ai = im[p0];
      float br = re[p1], bi = im[p1];
      float tr = br * wr - bi * wi;
      float ti = br * wi + bi * wr;
      re[p0] = ar + tr; im[p0] = ai + ti;
      re[p1] = ar - tr; im[p1] = ai - ti;
    }
    __syncthreads();
  }

  // 16 rounds of block-wide argmax over bins 0..4096 (ties -> lower index)
  for (int round = 0; round < 16; ++round) {
    float bv = -1.0f;
    int bi2 = 0x7fffffff;
    for (int k = tid; k <= 4096; k += 256) {
      bool used = false;
      for (int r = 0; r < round; ++r) used |= (selk[r] == k);
      if (!used) {
        float m2 = re[k] * re[k] + im[k] * im[k];
        if (m2 > bv) { bv = m2; bi2 = k; }
      }
    }
    redv[tid] = bv;
    redi[tid] = bi2;
    __syncthreads();
    for (int off = 128; off > 0; off >>= 1) {
      if (tid < off) {
        float v2 = redv[tid + off];
        int i2 = redi[tid + off];
        if (v2 > redv[tid] || (v2 == redv[tid] && i2 < redi[tid])) {
          redv[tid] = v2;
          redi[tid] = i2;
        }
      }
      __syncthreads();
    }
    if (tid == 0) {
      int k = redi[0];
      selk[round] = k;
      selR[round] = re[k];
      selI[round] = im[k];
      selW[round] = (k == 0 || k == 4096) ? 1.0f : 2.0f;
    }
    __syncthreads();
  }

  // sparse irfft: out[t] = (1/N) * sum_k w_k*(Re_k cos - Im_k sin), q=k*t mod N
  for (int t = tid; t < 8192; t += 256) {
    float acc = 0.0f;
#pragma unroll
    for (int r = 0; r < 16; ++r) {
      int q = (selk[r] * t) & 8191;
      float c, sn;
      if (q < 4096) { c = twr[q]; sn = -twi[q]; }
      else          { c = -twr[q - 4096]; sn = twi[q - 4096]; }
      acc += selW[r] * (selR[r] * c - selI[r] * sn);
    }
    fsC[sig + t] = acc * (1.0f / 8192.0f);
  }
}

// ---------------------------------------------------------------------------
// Kernel 3: out = fs @ W + b ; seasonal = LN(seasonal + out).
// 64x128 tile per workgroup, f32 WMMA 16x16x4, wave-per-row LayerNorm.
// ---------------------------------------------------------------------------
__global__ __launch_bounds__(256) void gemm_ln_kernel(
    const float* __restrict__ fsC, float* __restrict__ seasR,
    float* __restrict__ seasC, const float* __restrict__ Wg,
    const float* __restrict__ bg, const float* __restrict__ gamma,
    const float* __restrict__ beta) {
  extern __shared__ float sm[];
  float* Wls = sm;               // 128 x 136 (padded: half-waves hit disjoint banks)
  float* Als = sm + 128 * 136;   // A as [k=128][m=64], pitch 72; reused as S[64][129]
  float* S = Als;
  const int tid = threadIdx.x;
  const int lane = tid & 31, wv = tid >> 5;
  const int hf = lane >> 4, ln = lane & 15;
  const int l0 = blockIdx.x * 64;
  const int b = blockIdx.y;

  for (int i = tid; i < 128 * 128; i += 256) {
    int k = i >> 7, n = i & 127;
    Wls[k * 136 + n] = Wg[i];
  }
  for (int i = tid; i < 64 * 128; i += 256) {
    int d = i >> 6, l = i & 63;
    Als[d * 72 + l] = fsC[((size_t)b * DCH + d) * LTOT + l0 + l];
  }
  __syncthreads();

  const int n0 = wv * 16;               // 8 waves cover N = 0..127
  const float biasv = bg[n0 + ln];
  v8f accs[4];
  for (int mb = 0; mb < 4; ++mb) {      // 4 M-blocks of 16 rows
    v8f acc = {0.f, 0.f, 0.f, 0.f, 0.f, 0.f, 0.f, 0.f};
#pragma unroll 4
    for (int k0 = 0; k0 < 128; k0 += 4) {
      int ka = k0 + 2 * hf;             // lanes 0-15: K=k0,k0+1; 16-31: K=k0+2,k0+3
      v2f av;
      av.x = Als[ka * 72 + mb * 16 + ln];
      av.y = Als[(ka + 1) * 72 + mb * 16 + ln];
      v2f bv;
      bv.x = Wls[ka * 136 + n0 + ln];
      bv.y = Wls[(ka + 1) * 136 + n0 + ln];
      acc = __builtin_amdgcn_wmma_f32_16x16x4_f32(
          false, av, false, bv, (short)0, acc, false, false);
    }
    accs[mb] = acc;
  }
  __syncthreads();  // done reading Als; reuse as S

  // D-matrix layout: VGPR r holds M=r (lanes 0-15) and M=r+8 (lanes 16-31)
  for (int mb = 0; mb < 4; ++mb) {
#pragma unroll
    for (int r = 0; r < 8; ++r) {
      int row = mb * 16 + hf * 8 + r;
      int col = n0 + ln;
      float resv = seasR[((size_t)b * LTOT + l0 + row) * DCH + col];
      S[row * 129 + col] = accs[mb][r] + biasv + resv;
    }
  }
  __syncthreads();

  // wave-per-row LayerNorm over D=128
  for (int row = wv; row < 64; row += 8) {
    float vv[4];
#pragma unroll
    for (int i = 0; i < 4; ++i) vv[i] = S[row * 129 + lane * 4 + i];
    float s = vv[0] + vv[1] + vv[2] + vv[3];
    s = waveRedSum(s);
    float mean = s * (1.0f / 128.0f);
    float sq = 0.f;
#pragma unroll
    for (int i = 0; i < 4; ++i) { float dlt = vv[i] - mean; sq += dlt * dlt; }
    sq = waveRedSum(sq);
    float rs = rsqrtf(sq * (1.0f / 128.0f) + 1e-6f);
#pragma unroll
    for (int i = 0; i < 4; ++i) {
      int d = lane * 4 + i;
      S[row * 129 + d] = (vv[i] - mean) * rs * gamma[d] + beta[d];
    }
  }
  __syncthreads();

  for (int i = tid; i < 8192; i += 256) {
    int l = i >> 7, d = i & 127;
    seasR[((size_t)b * LTOT + l0 + l) * DCH + d] = S[l * 129 + d];
  }
  for (int i = tid; i < 8192; i += 256) {
    int d = i >> 6, l = i & 63;
    seasC[((size_t)b * DCH + d) * LTOT + l0 + l] = S[l * 129 + d];
  }
}

// ---------------------------------------------------------------------------
// Kernel 4: out = LN(seasonal + trend), wave per row.
// ---------------------------------------------------------------------------
__global__ __launch_bounds__(256) void final_ln_kernel(
    const float* __restrict__ seasR, const float* __restrict__ trend,
    const float* __restrict__ gamma, const float* __restrict__ beta,
    float* __restrict__ out) {
  const int lane = threadIdx.x & 31, wv = threadIdx.x >> 5;
  const size_t row = (size_t)blockIdx.x * 8 + wv;
  const size_t base = row * DCH;
  float vv[4];
#pragma unroll
  for (int i = 0; i < 4; ++i) {
    int d = lane * 4 + i;
    vv[i] = seasR[base + d] + trend[base + d];
  }
  float s = vv[0] + vv[1] + vv[2] + vv[3];
  s = waveRedSum(s);
  float mean = s * (1.0f / 128.0f);
  float sq = 0.f;
#pragma unroll
  for (int i = 0; i < 4; ++i) { float dlt = vv[i] - mean; sq += dlt * dlt; }
  sq = waveRedSum(sq);
  float rs = rsqrtf(sq * (1.0f / 128.0f) + 1e-6f);
#pragma unroll
  for (int i = 0; i < 4; ++i) {
    int d = lane * 4 + i;
    out[base + d] = (vv[i] - mean) * rs * gamma[d] + beta[d];
  }
}

// ---------------------------------------------------------------------------
extern "C" void kernel_launch(void* const* d_in, const int* in_sizes, int n_in,
                              void* d_out, int out_size, void* d_ws,
                              size_t ws_size, hipStream_t stream) {
  (void)in_sizes; (void)n_in; (void)out_size; (void)ws_size;
  const float* x   = (const float*)d_in[0];
  const float* dw  = (const float*)d_in[1];
  const float* db  = (const float*)d_in[2];
  const float* lns = (const float*)d_in[3];
  const float* lnb = (const float*)d_in[4];
  const float* fsc = (const float*)d_in[5];
  const float* fbi = (const float*)d_in[6];
  float* out = (float*)d_out;

  const size_t NTOT = (size_t)32 * LTOT * DCH;  // 33,554,432 floats
  float* ws = (float*)d_ws;
  float* trend = ws;              // row-major
  float* seasR = ws + NTOT;       // row-major seasonal
  float* seasC = ws + 2 * NTOT;   // channel-major seasonal [B,D,L]
  float* fsC   = ws + 3 * NTOT;   // channel-major fourier output

  const size_t smDecomp  = (size_t)(88 * 128 + 64 * 129) * sizeof(float);   // 78 KB
  const size_t smFourier = (size_t)(2 * 8192 + 2 * 4096) * sizeof(float);   // 96 KB
  const size_t smGemm    = (size_t)(128 * 136 + 128 * 72) * sizeof(float);  // 104 KB

  decomp_kernel<<<dim3(LTOT / 64, 32), 256, smDecomp, stream>>>(
      x, trend, seasR, seasC);

  for (int layer = 0; layer < 2; ++layer) {
    fourier_kernel<<<dim3(32 * DCH), 256, smFourier, stream>>>(seasC, fsC);
    gemm_ln_kernel<<<dim3(LTOT / 64, 32), 256, smGemm, stream>>>(
        fsC, seasR, seasC, dw + (size_t)layer * DCH * DCH,
        db + (size_t)layer * DCH, lns + (size_t)layer * DCH,
        lnb + (size_t)layer * DCH);
  }

  final_ln_kernel<<<dim3((32 * LTOT) / 8), 256, 0, stream>>>(
      seasR, trend, fsc, fbi, out);
}